// ColorCNN_30545807409602
// MI455X (gfx1250) — compile-verified
//
#include <hip/hip_runtime.h>
#include <math.h>

// Problem constants (match reference)
constexpr int  kB   = 4;
constexpr int  kH   = 192;
constexpr int  kW   = 192;
constexpr long kHW  = (long)kH * kW;        // 36864
constexpr long kNP  = (long)kB * kHW;       // 147456 pixels
constexpr int  kCIN = 64;
constexpr int  kCB  = 16;
constexpr int  kCM  = 256;
constexpr float kEPS = 1e-5f;

typedef __attribute__((ext_vector_type(2))) float v2f;
typedef __attribute__((ext_vector_type(8))) float v8f;

// Workspace layout (float offsets)
constexpr long WS_SUM   = 0;                 // 16
constexpr long WS_SUMSQ = 16;                // 16
constexpr long WS_SCALE = 32;                // 16
constexpr long WS_SHIFT = 48;                // 16
constexpr long WS_NUM   = 64;                // B*3*256 = 3072   [b][c][k]
constexpr long WS_DEN   = WS_NUM + 3072;     // B*256  = 1024    [b][k]
constexpr long WS_X     = 4352;              // NP*16 (pixel-major bottleneck pre-BN)
constexpr long WS_SM    = WS_X + kNP * 16;   // NP*16 softmax planes [s][pix]
constexpr long WS_IDX   = WS_SM + kNP * 16;  // NP*16 index planes   [s][pix] (int)

// ---------------------------------------------------------------------------
// k1: bottleneck GEMM  x[pix][16] = base_feat[pix][64] * w_bneck^T + b
//     fp32 WMMA 16x16x4, K=64 -> 16 steps; each wave does two 16-pixel tiles.
//     Also accumulates per-channel sum / sum-of-squares for BatchNorm.
// ---------------------------------------------------------------------------
__global__ __launch_bounds__(256) void k1_bneck(const float* __restrict__ bf,
                                                const float* __restrict__ wb,
                                                const float* __restrict__ bb,
                                                float* __restrict__ ws) {
  const int tid  = threadIdx.x;
  const int lane = tid & 31;
  const int m    = lane & 15;
  const int half = lane >> 4;
  const int wave = (blockIdx.x * 256 + tid) >> 5;     // 0..4607
  const long pix0 = (long)wave * 32;                  // global pixel base
  const int  b    = (int)(pix0 / kHW);
  const long pBm  = pix0 - (long)b * kHW + m;         // in-image pixel for row m
  const float* __restrict__ base = bf + (long)b * kCIN * kHW;

  const float bias = bb[m];                            // lane's output channel
  v8f c0, c1;
#pragma unroll
  for (int v = 0; v < 8; ++v) { c0[v] = bias; c1[v] = bias; }

#pragma unroll 4
  for (int kk = 0; kk < 16; ++kk) {
    const int kb = kk * 4 + half * 2;                  // A/B K pair per lane half
    v2f bv;
    { const float2 t = *(const float2*)(wb + m * kCIN + kb); bv.x = t.x; bv.y = t.y; }
    v2f a0, a1;
    a0.x = base[(long)kb * kHW + pBm];
    a0.y = base[(long)(kb + 1) * kHW + pBm];
    a1.x = base[(long)kb * kHW + pBm + 16];
    a1.y = base[(long)(kb + 1) * kHW + pBm + 16];
    c0 = __builtin_amdgcn_wmma_f32_16x16x4_f32(false, a0, false, bv, (short)0, c0, false, false);
    c1 = __builtin_amdgcn_wmma_f32_16x16x4_f32(false, a1, false, bv, (short)0, c1, false, false);
  }

  float* __restrict__ x = ws + WS_X;
  float s = 0.f, sq = 0.f;
#pragma unroll
  for (int v = 0; v < 8; ++v) {
    const int r = v + 8 * half;                        // D row -> pixel offset
    x[(pix0 + r) * 16 + m]      = c0[v];               // coalesced 64B per half
    x[(pix0 + 16 + r) * 16 + m] = c1[v];
    s  += c0[v] + c1[v];
    sq += c0[v] * c0[v] + c1[v] * c1[v];
  }
  atomicAdd(ws + WS_SUM + m, s);
  atomicAdd(ws + WS_SUMSQ + m, sq);
}

// ---------------------------------------------------------------------------
// k2: fold batch stats into fused BN affine: scale = gamma*rsqrt(var+eps),
//     shift = beta - mean*scale
// ---------------------------------------------------------------------------
__global__ void k2_stats(const float* __restrict__ gam, const float* __restrict__ bet,
                         float* __restrict__ ws) {
  const int ch = threadIdx.x;
  if (ch < 16) {
    const float N    = (float)kNP;
    const float mean = ws[WS_SUM + ch] / N;
    const float var  = ws[WS_SUMSQ + ch] / N - mean * mean;
    const float sc   = gam[ch] * rsqrtf(var + kEPS);
    ws[WS_SCALE + ch] = sc;
    ws[WS_SHIFT + ch] = bet[ch] - mean * sc;
  }
}

// ---------------------------------------------------------------------------
// k3: fused BN+ReLU -> mask GEMM (WMMA fp32, 16 color tiles x 4 K-steps)
//     -> incremental top-16 via LDS transpose -> softmax -> sparse mask write
//     -> topk planes + LDS-aggregated num/den accumulation.
// ---------------------------------------------------------------------------
__global__ __launch_bounds__(256) void k3_mask(const float* __restrict__ img,
                                               const float* __restrict__ wm,
                                               float* __restrict__ ws,
                                               float* __restrict__ out_msp) {
  __shared__ float tileS[8][32][17];   // per-wave 32-pixel x 16-color transpose tile
  __shared__ float denL[256];
  __shared__ float numL[3 * 256];

  const int tid  = threadIdx.x;
  const int lane = tid & 31;
  const int wave = tid >> 5;
  const int m    = lane & 15;
  const int half = lane >> 4;

  denL[tid] = 0.f; numL[tid] = 0.f; numL[256 + tid] = 0.f; numL[512 + tid] = 0.f;
  __syncthreads();

  const long blkPix  = (long)blockIdx.x * 256;
  const int  b       = (int)(blkPix / kHW);            // 36864 % 256 == 0
  const long wavePix = blkPix + (long)wave * 32;

  const float* __restrict__ x  = ws + WS_X;
  const float* __restrict__ sc = ws + WS_SCALE;
  const float* __restrict__ sh = ws + WS_SHIFT;

  // A fragments (BN+ReLU applied) for both 16-pixel tiles, all 4 K-steps
  v2f a0[4], a1[4];
#pragma unroll
  for (int kk = 0; kk < 4; ++kk) {
    const int kb = kk * 4 + half * 2;
    const float s0 = sc[kb], s1 = sc[kb + 1];
    const float h0 = sh[kb], h1 = sh[kb + 1];
    const float2 t0 = *(const float2*)(x + (wavePix + m) * 16 + kb);
    const float2 t1 = *(const float2*)(x + (wavePix + 16 + m) * 16 + kb);
    a0[kk].x = fmaxf(t0.x * s0 + h0, 0.f);
    a0[kk].y = fmaxf(t0.y * s1 + h1, 0.f);
    a1[kk].x = fmaxf(t1.x * s0 + h0, 0.f);
    a1[kk].y = fmaxf(t1.y * s1 + h1, 0.f);
  }

  float tv[16]; int ti[16];
#pragma unroll
  for (int s = 0; s < 16; ++s) { tv[s] = -3.0e38f; ti[s] = 0; }
  float mn = -3.0e38f;

  float (*T)[17] = tileS[wave];

  for (int ct = 0; ct < 16; ++ct) {
    v8f c0, c1;
#pragma unroll
    for (int v = 0; v < 8; ++v) { c0[v] = 0.f; c1[v] = 0.f; }
#pragma unroll
    for (int kk = 0; kk < 4; ++kk) {
      const int kb = kk * 4 + half * 2;
      v2f bv;
      { const float2 t = *(const float2*)(wm + (ct * 16 + m) * kCB + kb); bv.x = t.x; bv.y = t.y; }
      c0 = __builtin_amdgcn_wmma_f32_16x16x4_f32(false, a0[kk], false, bv, (short)0, c0, false, false);
      c1 = __builtin_amdgcn_wmma_f32_16x16x4_f32(false, a1[kk], false, bv, (short)0, c1, false, false);
    }
#pragma unroll
    for (int v = 0; v < 8; ++v) {
      const int r = v + 8 * half;
      T[r][m]      = c0[v];
      T[16 + r][m] = c1[v];
    }
    __syncthreads();
    // lane owns pixel (wavePix + lane): scan its 16 new color scores
#pragma unroll
    for (int j = 0; j < 16; ++j) {
      float val = T[lane][j];
      if (val > mn) {                                  // guarded bubble-insert
        float cv = val; int ci = ct * 16 + j;
#pragma unroll
        for (int s = 0; s < 16; ++s) {
          const bool gt = cv > tv[s];
          const float ov = tv[s]; const int oi = ti[s];
          tv[s] = gt ? cv : ov;  ti[s] = gt ? ci : oi;
          cv    = gt ? ov : cv;  ci    = gt ? oi : ci;
        }
        mn = tv[15];
      }
    }
    __syncthreads();
  }

  // softmax over the (descending) top-16
  float e[16]; float ssum = 0.f;
  const float mx = tv[0];
#pragma unroll
  for (int s = 0; s < 16; ++s) { e[s] = expf(tv[s] - mx); ssum += e[s]; }
  const float isum = 1.f / ssum;
#pragma unroll
  for (int s = 0; s < 16; ++s) e[s] *= isum;

  const long pg = wavePix + lane;                      // this thread's pixel
  const long pi = pg - (long)b * kHW;

  // sparse mask [B,256,H,W]: zero the column, then scatter (stores stay ordered)
  float* __restrict__ msp = out_msp + (long)b * kCM * kHW + pi;
#pragma unroll 8
  for (int ch = 0; ch < 256; ++ch) msp[(long)ch * kHW] = 0.f;
#pragma unroll
  for (int s = 0; s < 16; ++s) msp[(long)ti[s] * kHW] = e[s];

  // stash top-k planes for the recolor pass
  float* __restrict__ smp = ws + WS_SM;
  int*   __restrict__ idp = (int*)(ws + WS_IDX);
#pragma unroll
  for (int s = 0; s < 16; ++s) {
    smp[(long)s * kNP + pg] = e[s];
    idp[(long)s * kNP + pg] = ti[s];
  }

  // num/den accumulation, LDS-aggregated per block (whole block same image b)
  const float i0 = img[((long)b * 3 + 0) * kHW + pi];
  const float i1 = img[((long)b * 3 + 1) * kHW + pi];
  const float i2 = img[((long)b * 3 + 2) * kHW + pi];
#pragma unroll
  for (int s = 0; s < 16; ++s) {
    atomicAdd(&denL[ti[s]], e[s]);
    atomicAdd(&numL[ti[s]],        i0 * e[s]);
    atomicAdd(&numL[256 + ti[s]],  i1 * e[s]);
    atomicAdd(&numL[512 + ti[s]],  i2 * e[s]);
  }
  __syncthreads();
  if (denL[tid] != 0.f) atomicAdd(ws + WS_DEN + b * 256 + tid, denL[tid]);
#pragma unroll
  for (int c = 0; c < 3; ++c) {
    const float v = numL[c * 256 + tid];
    if (v != 0.f) atomicAdd(ws + WS_NUM + ((long)b * 3 + c) * 256 + tid, v);
  }
}

// ---------------------------------------------------------------------------
// k4: palette[b][c][k] = num/(den+1e-8)   (COLOR_NORM == 1)
// ---------------------------------------------------------------------------
__global__ void k4_pal(const float* __restrict__ ws, float* __restrict__ pal) {
  const int i = blockIdx.x * blockDim.x + threadIdx.x;  // flat [b][c][k]
  if (i < kB * 3 * 256) {
    const int bb = i / 768;
    const int k  = i & 255;
    pal[i] = ws[WS_NUM + i] / (ws[WS_DEN + bb * 256 + k] + 1e-8f);
  }
}

// ---------------------------------------------------------------------------
// k5: transformed[b][c][pix] = sum_s sm[s] * palette[b][c][idx[s]]
// ---------------------------------------------------------------------------
__global__ __launch_bounds__(256) void k5_recolor(const float* __restrict__ ws,
                                                  const float* __restrict__ pal,
                                                  float* __restrict__ out) {
  const long pg = (long)blockIdx.x * 256 + threadIdx.x;
  const int  b  = (int)(pg / kHW);
  const long pi = pg - (long)b * kHW;
  const float* __restrict__ smp = ws + WS_SM;
  const int*   __restrict__ idp = (const int*)(ws + WS_IDX);
  const float* __restrict__ p0 = pal + ((long)b * 3 + 0) * 256;
  const float* __restrict__ p1 = pal + ((long)b * 3 + 1) * 256;
  const float* __restrict__ p2 = pal + ((long)b * 3 + 2) * 256;
  float a0 = 0.f, a1 = 0.f, a2 = 0.f;
#pragma unroll
  for (int s = 0; s < 16; ++s) {
    const float w = smp[(long)s * kNP + pg];
    const int  id = idp[(long)s * kNP + pg];
    a0 += w * p0[id]; a1 += w * p1[id]; a2 += w * p2[id];
  }
  out[((long)b * 3 + 0) * kHW + pi] = a0;
  out[((long)b * 3 + 1) * kHW + pi] = a1;
  out[((long)b * 3 + 2) * kHW + pi] = a2;
}

// ---------------------------------------------------------------------------
extern "C" void kernel_launch(void* const* d_in, const int* in_sizes, int n_in,
                              void* d_out, int out_size, void* d_ws, size_t ws_size,
                              hipStream_t stream) {
  const float* img = (const float*)d_in[0];
  const float* bf  = (const float*)d_in[1];
  const float* wb  = (const float*)d_in[2];
  const float* bb  = (const float*)d_in[3];
  const float* gam = (const float*)d_in[4];
  const float* bet = (const float*)d_in[5];
  const float* wm  = (const float*)d_in[6];
  (void)in_sizes; (void)n_in; (void)out_size; (void)ws_size;

  float* ws      = (float*)d_ws;
  float* out_img = (float*)d_out;                         // [B,3,H,W]
  float* out_msp = out_img + (long)kB * 3 * kHW;          // [B,256,H,W]
  float* out_pal = out_msp + (long)kB * kCM * kHW;        // [B,3,256]

  hipMemsetAsync(ws + WS_SUM, 0, 32 * sizeof(float), stream);
  hipMemsetAsync(ws + WS_NUM, 0, (3072 + 1024) * sizeof(float), stream);

  const int nblk = (int)(kNP / 256);                      // 576
  k1_bneck<<<nblk, 256, 0, stream>>>(bf, wb, bb, ws);
  k2_stats<<<1, 32, 0, stream>>>(gam, bet, ws);
  k3_mask<<<nblk, 256, 0, stream>>>(img, wm, ws, out_msp);
  k4_pal<<<(kB * 3 * 256 + 255) / 256, 256, 0, stream>>>(ws, out_pal);
  k5_recolor<<<nblk, 256, 0, stream>>>(ws, out_pal, out_img);
}